// GCNModel_37512244363809
// MI455X (gfx1250) — compile-verified
//
#include <hip/hip_runtime.h>
#include <math.h>

typedef __attribute__((ext_vector_type(2)))  float    v2f;
typedef __attribute__((ext_vector_type(8)))  float    v8f;

#define HDIM 128
#define WT_STRIDE 130   // transposed-W LDS row stride (even -> b64 aligned; no per-wave bank conflicts)

// ---------------- degree / normalization ----------------
__global__ void k_init_ones(float* deg, int n) {
  int i = blockIdx.x * blockDim.x + threadIdx.x;
  if (i < n) deg[i] = 1.0f;                 // deg = 1 (self loop)
}

__global__ void k_count(const int* __restrict__ dst, float* deg, int e) {
  int i = blockIdx.x * blockDim.x + threadIdx.x;
  if (i < e) atomicAdd(&deg[dst[i]], 1.0f);
}

__global__ void k_norm(float* di, float* dis, int n) {
  int i = blockIdx.x * blockDim.x + threadIdx.x;
  if (i < n) {
    float d = di[i];
    dis[i] = rsqrtf(d);                     // deg^-1/2
    di[i]  = 1.0f / d;                      // deg^-1 (self-loop coef)
  }
}

// ---------------- WMMA GEMM: Out[n x 128] = A[n x 128] @ W[128 x 128] ----------------
// Block = 4 waves; W staged transposed in LDS once per block; one wave -> 16x128 output.
__global__ void __launch_bounds__(128) k_gemm128(const float* __restrict__ A,
                                                 const float* __restrict__ W,
                                                 float* __restrict__ Out, int n) {
  __shared__ float lw[HDIM * WT_STRIDE];    // W^T: lw[col*WT_STRIDE + k] = W[k*128 + col]

  // cooperative staged transpose: coalesced float4 global reads
  for (int i4 = threadIdx.x; i4 < HDIM * (HDIM / 4); i4 += 128) {
    int k  = i4 >> 5;                       // row of W
    int c4 = (i4 & 31) << 2;                // starting col
    float4 w4 = reinterpret_cast<const float4*>(W)[i4];
    lw[(c4 + 0) * WT_STRIDE + k] = w4.x;
    lw[(c4 + 1) * WT_STRIDE + k] = w4.y;
    lw[(c4 + 2) * WT_STRIDE + k] = w4.z;
    lw[(c4 + 3) * WT_STRIDE + k] = w4.w;
  }
  __syncthreads();

  int wave = threadIdx.x >> 5;
  int lane = threadIdx.x & 31;
  int ntiles = (n + 15) >> 4;
  int tile = blockIdx.x * 4 + wave;
  if (tile >= ntiles) return;               // after barrier: safe

  int rowBase = tile << 4;
  int r  = lane & 15;
  int hi = lane >> 4;                       // half-wave select
  int arow = rowBase + r; if (arow > n - 1) arow = n - 1;
  const float* Ap = A + (size_t)arow * HDIM;

  v8f acc[8] = {};

  // V_WMMA_F32_16X16X4_F32 (exact f32).
  // A 16x4: lanes0-15 K={k,k+1}, lanes16-31 K={k+2,k+3}.
  // B 4x16: VGPR v holds K = k + 2*hi + v, N = lane&15 -> one ds_load_b64 from W^T.
  for (int k0 = 0; k0 < HDIM; k0 += 4) {
    int ka = k0 + hi * 2;
    v2f a = *reinterpret_cast<const v2f*>(Ap + ka);   // global_load_b64
#pragma unroll
    for (int t = 0; t < 8; ++t) {
      int col = t * 16 + r;
      v2f b = *reinterpret_cast<const v2f*>(lw + col * WT_STRIDE + ka);  // ds_load_b64
      acc[t] = __builtin_amdgcn_wmma_f32_16x16x4_f32(false, a, false, b,
                                                     (short)0, acc[t], false, false);
    }
  }

  // C/D layout: VGPR v -> row = rowBase + v + 8*hi, col = t*16 + (lane&15)
  if (rowBase + 16 <= n) {                  // uniform fast path: no per-store guards
#pragma unroll
    for (int t = 0; t < 8; ++t) {
#pragma unroll
      for (int v = 0; v < 8; ++v) {
        Out[(size_t)(rowBase + v + hi * 8) * HDIM + t * 16 + r] = acc[t][v];
      }
    }
  } else {
#pragma unroll
    for (int t = 0; t < 8; ++t) {
#pragma unroll
      for (int v = 0; v < 8; ++v) {
        int row = rowBase + v + hi * 8;
        if (row < n) Out[(size_t)row * HDIM + t * 16 + r] = acc[t][v];
      }
    }
  }
}

// ---------------- self-loop + bias init: agg = h * di + b ----------------
__global__ void k_init_agg(const float* __restrict__ h, float* __restrict__ agg,
                           const float* __restrict__ di, const float* __restrict__ b, int n) {
  size_t i = (size_t)blockIdx.x * blockDim.x + threadIdx.x;
  if (i >= (size_t)n * HDIM) return;
  int row = (int)(i >> 7);
  int c   = (int)(i & 127);
  agg[i] = h[i] * di[row] + b[c];
}

// ---------------- edge scatter: one wave per edge, float4 gather + f32 atomics ----------------
__global__ void k_scatter(const float* __restrict__ h, float* __restrict__ agg,
                          const int* __restrict__ src, const int* __restrict__ dst,
                          const float* __restrict__ dis, int e) {
  int w = blockIdx.x * (blockDim.x >> 5) + (threadIdx.x >> 5);
  if (w >= e) return;
  int lane = threadIdx.x & 31;
  int s = src[w], d = dst[w];
  float coef = dis[s] * dis[d];
  float4 v = reinterpret_cast<const float4*>(h + (size_t)s * HDIM)[lane];
  float* ap = agg + (size_t)d * HDIM + lane * 4;
  atomicAdd(ap + 0, v.x * coef);
  atomicAdd(ap + 1, v.y * coef);
  atomicAdd(ap + 2, v.z * coef);
  atomicAdd(ap + 3, v.w * coef);
}

// ---------------- batch-norm statistics ----------------
__global__ void k_zero_stats(float* sums, float* sqs) {
  if (threadIdx.x < HDIM) { sums[threadIdx.x] = 0.f; sqs[threadIdx.x] = 0.f; }
}

__global__ void k_stats(const float* __restrict__ a, float* sums, float* sqs, int n) {
  __shared__ float sh[256], sh2[256];
  float s = 0.f, s2 = 0.f;
  size_t total  = (size_t)n * HDIM;
  size_t stride = (size_t)gridDim.x * blockDim.x;   // multiple of 128 -> fixed column per thread
  for (size_t i = (size_t)blockIdx.x * blockDim.x + threadIdx.x; i < total; i += stride) {
    float v = a[i]; s += v; s2 += v * v;
  }
  sh[threadIdx.x] = s; sh2[threadIdx.x] = s2;
  __syncthreads();
  if (threadIdx.x < HDIM) {
    atomicAdd(&sums[threadIdx.x], sh[threadIdx.x]  + sh[threadIdx.x + HDIM]);
    atomicAdd(&sqs[threadIdx.x],  sh2[threadIdx.x] + sh2[threadIdx.x + HDIM]);
  }
}

__global__ void k_bn_relu(const float* __restrict__ a, float* __restrict__ out,
                          const float* __restrict__ sums, const float* __restrict__ sqs,
                          const float* __restrict__ g, const float* __restrict__ be, int n) {
  size_t i = (size_t)blockIdx.x * blockDim.x + threadIdx.x;
  if (i >= (size_t)n * HDIM) return;
  int c = (int)(i & 127);
  float inv = 1.0f / (float)n;
  float mu  = sums[c] * inv;
  float var = sqs[c] * inv - mu * mu;       // population variance (ddof=0)
  float v = (a[i] - mu) * rsqrtf(var + 1e-5f) * g[c] + be[c];
  out[i] = v > 0.f ? v : 0.f;
}

// ---------------- layer 3: 128 -> 2 (wave per node, coalesced float4 + shuffle reduce) ----------------
__global__ void k_gemm_c2(const float* __restrict__ act, const float* __restrict__ W3,
                          float* __restrict__ h3, int n) {
  int w = blockIdx.x * (blockDim.x >> 5) + (threadIdx.x >> 5);
  if (w >= n) return;
  int lane = threadIdx.x & 31;
  float4 v = reinterpret_cast<const float4*>(act + (size_t)w * HDIM)[lane];
  int k4 = lane * 4;
  float s0 = v.x * W3[(k4 + 0) * 2] + v.y * W3[(k4 + 1) * 2]
           + v.z * W3[(k4 + 2) * 2] + v.w * W3[(k4 + 3) * 2];
  float s1 = v.x * W3[(k4 + 0) * 2 + 1] + v.y * W3[(k4 + 1) * 2 + 1]
           + v.z * W3[(k4 + 2) * 2 + 1] + v.w * W3[(k4 + 3) * 2 + 1];
#pragma unroll
  for (int off = 16; off > 0; off >>= 1) {
    s0 += __shfl_down(s0, off);
    s1 += __shfl_down(s1, off);
  }
  if (lane == 0) {
    h3[w * 2 + 0] = s0;
    h3[w * 2 + 1] = s1;
  }
}

__global__ void k_init_agg2(const float* __restrict__ h3, float* __restrict__ out,
                            const float* __restrict__ di, const float* __restrict__ b3, int n) {
  int i = blockIdx.x * blockDim.x + threadIdx.x;
  if (i >= 2 * n) return;
  int row = i >> 1, c = i & 1;
  out[i] = h3[i] * di[row] + b3[c];
}

__global__ void k_scatter2(const float* __restrict__ h3, float* __restrict__ out,
                           const int* __restrict__ src, const int* __restrict__ dst,
                           const float* __restrict__ dis, int e) {
  int i = blockIdx.x * blockDim.x + threadIdx.x;
  if (i >= e) return;
  int s = src[i], d = dst[i];
  float coef = dis[s] * dis[d];
  atomicAdd(&out[d * 2 + 0], h3[s * 2 + 0] * coef);
  atomicAdd(&out[d * 2 + 1], h3[s * 2 + 1] * coef);
}

__global__ void k_softmax2(float* out, int n) {
  int i = blockIdx.x * blockDim.x + threadIdx.x;
  if (i >= n) return;
  float a0 = out[i * 2], a1 = out[i * 2 + 1];
  float m  = fmaxf(a0, a1);
  float e0 = expf(a0 - m), e1 = expf(a1 - m);
  float inv = 1.0f / (e0 + e1);
  out[i * 2 + 0] = e0 * inv;
  out[i * 2 + 1] = e1 * inv;
}

// ---------------- launcher ----------------
extern "C" void kernel_launch(void* const* d_in, const int* in_sizes, int n_in,
                              void* d_out, int out_size, void* d_ws, size_t ws_size,
                              hipStream_t stream) {
  const float* x   = (const float*)d_in[0];
  const float* W1  = (const float*)d_in[1];
  const float* b1  = (const float*)d_in[2];
  const float* g1  = (const float*)d_in[3];
  const float* be1 = (const float*)d_in[4];
  const float* W2  = (const float*)d_in[5];
  const float* b2  = (const float*)d_in[6];
  const float* g2  = (const float*)d_in[7];
  const float* be2 = (const float*)d_in[8];
  const float* W3  = (const float*)d_in[9];
  const float* b3  = (const float*)d_in[10];
  const int*   ei  = (const int*)d_in[11];

  int n = in_sizes[0] / HDIM;     // 100000
  int e = in_sizes[11] / 2;       // 1600000
  const int* src = ei;
  const int* dst = ei + e;

  float* ws = (float*)d_ws;
  size_t NH = (size_t)n * HDIM;
  float* buf0 = ws;               // N*128
  float* buf1 = buf0 + NH;        // N*128
  float* dis  = buf1 + NH;        // N
  float* di   = dis + n;          // N (deg, then 1/deg)
  float* sums = di + n;           // 128
  float* sqs  = sums + HDIM;      // 128
  float* h3   = sqs + HDIM;       // 2*N
  float* out  = (float*)d_out;

  int gN    = (n + 255) / 256;
  int gE    = (e + 255) / 256;
  int gNH   = (int)((NH + 255) / 256);
  int gN2   = (2 * n + 255) / 256;
  int gGemm = (((n + 15) / 16) + 3) / 4;
  int gScat = (e + 7) / 8;        // 8 waves / block, 1 wave / edge
  int gC2   = (n + 7) / 8;        // 8 waves / block, 1 wave / node

  // degree -> dis = deg^-1/2, di = deg^-1
  k_init_ones<<<gN, 256, 0, stream>>>(di, n);
  k_count<<<gE, 256, 0, stream>>>(dst, di, e);
  k_norm<<<gN, 256, 0, stream>>>(di, dis, n);

  // ---- layer 1 ----
  k_gemm128<<<gGemm, 128, 0, stream>>>(x, W1, buf0, n);
  k_init_agg<<<gNH, 256, 0, stream>>>(buf0, buf1, di, b1, n);
  k_scatter<<<gScat, 256, 0, stream>>>(buf0, buf1, src, dst, dis, e);
  k_zero_stats<<<1, 256, 0, stream>>>(sums, sqs);
  k_stats<<<1024, 256, 0, stream>>>(buf1, sums, sqs, n);
  k_bn_relu<<<gNH, 256, 0, stream>>>(buf1, buf0, sums, sqs, g1, be1, n);

  // ---- layer 2 ----
  k_gemm128<<<gGemm, 128, 0, stream>>>(buf0, W2, buf1, n);
  k_init_agg<<<gNH, 256, 0, stream>>>(buf1, buf0, di, b2, n);
  k_scatter<<<gScat, 256, 0, stream>>>(buf1, buf0, src, dst, dis, e);
  k_zero_stats<<<1, 256, 0, stream>>>(sums, sqs);
  k_stats<<<1024, 256, 0, stream>>>(buf0, sums, sqs, n);
  k_bn_relu<<<gNH, 256, 0, stream>>>(buf0, buf1, sums, sqs, g2, be2, n);

  // ---- layer 3 + softmax ----
  k_gemm_c2<<<gC2, 256, 0, stream>>>(buf1, W3, h3, n);
  k_init_agg2<<<gN2, 256, 0, stream>>>(h3, out, di, b3, n);
  k_scatter2<<<gE, 256, 0, stream>>>(h3, out, src, dst, dis, e);
  k_softmax2<<<gN, 256, 0, stream>>>(out, n);
}